// AntialiasSampling_38156489457721
// MI455X (gfx1250) — compile-verified
//
#include <hip/hip_runtime.h>
#include <stdint.h>
#include <stddef.h>

typedef float v4f __attribute__((ext_vector_type(4)));
typedef int   v4i __attribute__((ext_vector_type(4)));

typedef __attribute__((address_space(1))) v4i* gas_v4i_ptr;
typedef __attribute__((address_space(3))) v4i* las_v4i_ptr;

namespace {
constexpr int N  = 8;
constexpr int H  = 256;
constexpr int W  = 256;
constexpr int C  = 64;          // channels (contiguous, NHWC)
constexpr int TS = 16;          // input pixels per tile side
constexpr int HT = TS + 2;      // tile + 1-pixel halo each side
constexpr int CG = C / 4;       // float4 channel groups = 16
constexpr int OH = 2 * H;
constexpr int OW = 2 * W;
constexpr int THREADS = 256;    // 8 waves (wave32)
}

// ---- CDNA5 async global->LDS copy (16B per lane), builtin or asm fallback ----
__device__ __forceinline__ void async_copy16(
    const __attribute__((address_space(1))) float* src,
    __attribute__((address_space(3))) float* dst) {
#if __has_builtin(__builtin_amdgcn_global_load_async_to_lds_b128)
  __builtin_amdgcn_global_load_async_to_lds_b128(
      (gas_v4i_ptr)src, (las_v4i_ptr)dst, 0, 0);
#else
  asm volatile("global_load_async_to_lds_b128 %0, %1, off"
               : : "v"(dst), "v"(src) : "memory");
#endif
}

__device__ __forceinline__ void wait_async0() {
#if __has_builtin(__builtin_amdgcn_s_wait_asynccnt)
  __builtin_amdgcn_s_wait_asynccnt(0);
#else
  asm volatile("s_wait_asynccnt 0" ::: "memory");
#endif
}

__global__ __launch_bounds__(THREADS)
void upfir2x_kernel(const float* __restrict__ xin, float* __restrict__ yout) {
  // 18 x 18 pixels x 64 ch fp32 = 82,944 B of LDS (3 WGs / WGP)
  __shared__ float tile[HT * HT * C];

  const int b   = blockIdx.z;
  const int r0  = blockIdx.y * TS;
  const int c0  = blockIdx.x * TS;
  const int tid = threadIdx.x;
  const int cg  = tid & (CG - 1);  // channel-group: fast dim across half-wave
  const int pg  = tid >> 4;        // pixel column within tile: 0..15

  const __attribute__((address_space(1))) float* gx =
      (const __attribute__((address_space(1))) float*)xin;
  __attribute__((address_space(3))) float* lt =
      (__attribute__((address_space(3))) float*)tile;

  // ---------------- Stage (18x18) x 64ch tile into LDS ----------------
  // unit = (pixel, channel-group); lanes 0..15 of a wave = 16 consecutive
  // cgroups of one pixel -> 256B contiguous global per half-wave,
  // 512B contiguous per wave32 async instruction.
  constexpr int UNITS = HT * HT * CG;  // 5184
  for (int u = tid; u < UNITS; u += THREADS) {
    const int pix = u >> 4;
    const int ucg = u & (CG - 1);
    const int tr  = pix / HT;
    const int tc  = pix - tr * HT;
    const int r   = r0 - 1 + tr;
    const int c   = c0 - 1 + tc;
    const int lofs = pix * C + ucg * 4;
    if ((unsigned)r < (unsigned)H && (unsigned)c < (unsigned)W) {
      const __attribute__((address_space(1))) float* gsrc =
          gx + ((((size_t)b * H + r) * W + c) * C + ucg * 4);
      async_copy16(gsrc, lt + lofs);
    } else {
      // zero halo outside the image (upfirdn zero padding)
      v4f z = {0.0f, 0.0f, 0.0f, 0.0f};
      *(v4f*)&tile[lofs] = z;
    }
  }
  wait_async0();     // own-wave async completion
  __syncthreads();   // cross-wave visibility

  // ---------------- Polyphase 2x2 upsample from LDS ----------------
  // per-dim taps [1,3,3,1]/4; even phase: 0.25*x[p-1]+0.75*x[p],
  // odd phase: 0.75*x[p]+0.25*x[p+1]; gain*factor^2 already folded in.
  const float A = 0.25f, Bw = 0.75f;
  const int   pc = pg;  // this thread's tile column

  // column-mix (even/odd output-column phases) for one halo row
  const float* colbase = &tile[pc * C + cg * 4];
  auto colmix = [&](int hr, v4f& ev, v4f& ov) {
    const float* bp = colbase + hr * (HT * C);
    v4f v0 = *(const v4f*)(bp);          // input col c-1
    v4f v1 = *(const v4f*)(bp + C);      // input col c
    v4f v2 = *(const v4f*)(bp + 2 * C);  // input col c+1
    ev = A * v0 + Bw * v1;
    ov = Bw * v1 + A * v2;
  };

  // rolling 3-row window of column-mixed values (2/3 rows reused per step)
  v4f e0, o0, e1, o1, e2, o2;
  colmix(0, e0, o0);
  colmix(1, e1, o1);

  const size_t orow_stride = (size_t)OW * C;     // one output row
  float* op = yout + ((((size_t)b * OH + 2 * r0) * OW + 2 * (c0 + pc)) * C + cg * 4);

#pragma unroll 4
  for (int row = 0; row < TS; ++row) {
    colmix(row + 2, e2, o2);                     // only the new halo row

    const v4f o00 = A * e0 + Bw * e1;            // (even row, even col)
    const v4f o01 = A * o0 + Bw * o1;            // (even row, odd  col)
    const v4f o10 = Bw * e1 + A * e2;            // (odd  row, even col)
    const v4f o11 = Bw * o1 + A * o2;            // (odd  row, odd  col)

    // streamed 537MB output: non-temporal, keep L2 for the input
    __builtin_nontemporal_store(o00, (v4f*)op);
    __builtin_nontemporal_store(o01, (v4f*)(op + C));
    __builtin_nontemporal_store(o10, (v4f*)(op + orow_stride));
    __builtin_nontemporal_store(o11, (v4f*)(op + orow_stride + C));
    op += 2 * orow_stride;

    e0 = e1; o0 = o1;                            // shift window
    e1 = e2; o1 = o2;
  }
}

extern "C" void kernel_launch(void* const* d_in, const int* in_sizes, int n_in,
                              void* d_out, int out_size, void* d_ws, size_t ws_size,
                              hipStream_t stream) {
  (void)in_sizes; (void)n_in; (void)out_size; (void)d_ws; (void)ws_size;
  const float* x = (const float*)d_in[0];
  float* y = (float*)d_out;
  dim3 grid(W / TS, H / TS, N);  // 16 x 16 x 8 = 2048 workgroups
  dim3 block(THREADS);
  hipLaunchKernelGGL(upfir2x_kernel, grid, block, 0, stream, x, y);
}